// one_var_28295244546159
// MI455X (gfx1250) — compile-verified
//
#include <hip/hip_runtime.h>
#include <cstdint>

// CDNA5 / gfx1250. wave32. V_WMMA_F32_16X16X4_F32 for the history GEMM
// (fp32 end-to-end: the autoregressive Bernoulli chain flips decisions under
// bf16-level noise), ds_bpermute shuffles for the serial chain.
//
// Round-1 fix: inner 16-step sampling loop is now ROLLED (only the 8-way
// register-block loop is unrolled) to eliminate the scratch spills seen in
// round 0; WMMA K-loop unrolled 4x for load/WMMA overlap.

typedef float v2f __attribute__((ext_vector_type(2)));
typedef float v8f __attribute__((ext_vector_type(8)));

#define TILE        128
#define LDS_STRIDE  129   // +1 pad -> bank-conflict-free column access
#define ROWS_PB     16    // batch rows per workgroup (WMMA M)

__global__ __launch_bounds__(256) void one_var_28295244546159_kernel(
    const float* __restrict__ weight,  // [N]
    const float* __restrict__ mask,    // [N][N] row-major, strictly lower tri
    const float* __restrict__ u,       // [B][N]
    float* __restrict__ sample,        // [B][N] out (also WMMA A operand)
    float* __restrict__ xhat,          // [B][N] out
    int N, int num_tiles)
{
  extern __shared__ float lds[];
  float* maskT = lds;                       // [TILE][LDS_STRIDE]: maskT[i][k] = mask[t0+k][t0+i]
  float* accT  = lds + TILE * LDS_STRIDE;   // [16][LDS_STRIDE]

  const int tid  = threadIdx.x;
  const int wv   = tid >> 5;        // wave 0..7
  const int lane = tid & 31;
  const int h    = lane >> 4;       // half-wave
  const int m    = lane & 15;
  const int b0   = blockIdx.x * ROWS_PB;
  const int rowL = 2 * wv + h;                       // local batch row for sampling
  const size_t rowG = (size_t)(b0 + rowL) * N;       // global row base (sampling)
  const size_t aRow = (size_t)(b0 + m) * N;          // WMMA A operand row (GEMM)

  for (int t = 0; t < num_tiles; ++t) {
    const int tile0 = t * TILE;
    __syncthreads();   // previous tile fully done before LDS reuse

    // ---- stage transposed mask diagonal block (global-coalesced read,
    //      stride-129 LDS write -> conflict-free)
    for (int e = tid; e < TILE * TILE; e += 256) {
      int k = e >> 7;          // mask row within tile
      int i = e & (TILE - 1);  // mask col within tile
      maskT[i * LDS_STRIDE + k] = mask[(size_t)(tile0 + k) * N + tile0 + i];
    }

    // ---- WMMA pull of history: acc[0:16][16wv..16wv+15] += S[:,0:tile0] * mask^T
    // fp32 WMMA: A 16x4 (K = vgpr + 2h, M = m), B 4x16 (K = vgpr + 2h, N = m)
    v8f D = {0.f, 0.f, 0.f, 0.f, 0.f, 0.f, 0.f, 0.f};
    const size_t bRow = (size_t)(tile0 + 16 * wv + m) * N;  // mask row = output col
    #pragma unroll 4
    for (int j0 = 0; j0 < tile0; j0 += 4) {
      v2f a = *(const v2f*)(sample + aRow + j0 + 2 * h);
      v2f b = *(const v2f*)(mask   + bRow + j0 + 2 * h);
      D = __builtin_amdgcn_wmma_f32_16x16x4_f32(false, a, false, b,
                                                (short)0, D, false, false);
    }
    // D layout: VGPR r -> row (r + 8h), col (16wv + m)
    #pragma unroll
    for (int r = 0; r < 8; ++r)
      accT[(r + 8 * h) * LDS_STRIDE + 16 * wv + m] = D[r];

    __syncthreads();

    // ---- per-lane sampling state: lane holds columns (m + 16j) of its row
    float acc_s[8], u_s[8], w_s[8];
    #pragma unroll
    for (int j = 0; j < 8; ++j) {
      acc_s[j] = accT[rowL * LDS_STRIDE + m + 16 * j];
      u_s[j]   = u[rowG + tile0 + m + 16 * j];
      w_s[j]   = weight[tile0 + m + 16 * j];
    }

    // ---- serial autoregressive chain: 128 steps, no barriers (wave-private).
    // Outer register-block loop unrolled (static acc_s[jr] indexing); inner
    // 16-step loop rolled to keep VGPR pressure low (no spills).
    #pragma unroll
    for (int jr = 0; jr < 8; ++jr) {
      #pragma unroll 1
      for (int i2 = 0; i2 < 16; ++i2) {
        const int i   = jr * 16 + i2;
        const int src = (lane & 16) | i2;
        float acc_i = __shfl(acc_s[jr], src, 32);
        float u_i   = __shfl(u_s[jr],   src, 32);
        float w_i   = __shfl(w_s[jr],   src, 32);
        // u < sigmoid(x)  <=>  u*(1 + exp(-x)) < 1   (rcp-free)
        float e  = __expf(-w_i * acc_i);
        float tv = fmaf(u_i, e, u_i);
        float s  = (tv < 1.0f) ? 1.0f : -1.0f;
        if (m == i2) sample[rowG + tile0 + i] = s;   // one lane per half-wave
        // rank-1 update of the tile (mask zeros above diagonal make j>=k no-ops)
        const float* mrow = maskT + i * LDS_STRIDE + m;
        #pragma unroll
        for (int j = 0; j < 8; ++j)
          acc_s[j] = fmaf(s, mrow[16 * j], acc_s[j]);
      }
    }

    // ---- x_hat for this tile: acc is final for these columns now
    // (strict lower-tri => no later sample touches them; last mask col == 0)
    #pragma unroll
    for (int j = 0; j < 8; ++j) {
      float e = __expf(-w_s[j] * acc_s[j]);
      xhat[rowG + tile0 + m + 16 * j] = 1.0f / (1.0f + e);
    }

    __threadfence_block();  // sample stores visible to next tile's WMMA A loads
  }
}

extern "C" void kernel_launch(void* const* d_in, const int* in_sizes, int n_in,
                              void* d_out, int out_size, void* d_ws, size_t ws_size,
                              hipStream_t stream) {
  (void)n_in; (void)d_ws; (void)ws_size; (void)out_size;
  const float* weight = (const float*)d_in[0];
  const float* mask   = (const float*)d_in[1];
  const float* u      = (const float*)d_in[2];
  const int N = in_sizes[0];
  const int B = in_sizes[2] / N;

  float* sample = (float*)d_out;
  float* xhat   = sample + (size_t)B * N;

  const int num_tiles = N / TILE;
  const size_t lds_bytes = (size_t)(TILE * LDS_STRIDE + ROWS_PB * LDS_STRIDE) * sizeof(float);
  // ~74.3 KB dynamic LDS (WGP has 320 KB) — opt in, ignore failure.
  (void)hipFuncSetAttribute((const void*)one_var_28295244546159_kernel,
                            hipFuncAttributeMaxDynamicSharedMemorySize,
                            (int)lds_bytes);

  one_var_28295244546159_kernel<<<dim3(B / ROWS_PB), dim3(256), lds_bytes, stream>>>(
      weight, mask, u, sample, xhat, N, num_tiles);
}